// GeneralDotProductAttention_22256520528802
// MI455X (gfx1250) — compile-verified
//
#include <hip/hip_runtime.h>

typedef float v2f __attribute__((ext_vector_type(2)));
typedef float v8f __attribute__((ext_vector_type(8)));

#define B_   32
#define N_   6
#define C_   512
#define H_   32
#define W_   32
#define QS_  256
#define KS_  256
#define CHW  (C_*H_*W_)          // 524288 floats per (b,n) plane
#define F4_PER_BATCH (CHW/4)     // 131072 float4 per output batch
#define BLOCKS_PER_BATCH 256     // 256 blocks * 256 thr * 2 f4 = 131072

// ---------------------------------------------------------------------------
// Kernel 1: query[b][kk] = sum_d q[b][d]*W[kk][d] + bias[kk]
// M=32 (batch), N=256 (KS), K=256 (QS).  V_WMMA_F32_16X16X4_F32, one wave
// per 16x16 output tile, K stepped by 4.  2 M-tiles x 16 N-tiles = 32 waves.
// A 16x4 f32 layout: lanes 0-15 -> M=lane, K={k0,k0+1}; lanes 16-31 -> K={k0+2,k0+3}
// B 4x16 f32 layout: lanes 0-15 -> N=lane, K={k0,k0+1}; lanes 16-31 -> K={k0+2,k0+3}
// C/D 16x16 f32: VGPR r, lane L: M = r + (L>=16 ? 8 : 0), N = L&15
// ---------------------------------------------------------------------------
__global__ void qproj_wmma(const float* __restrict__ q,
                           const float* __restrict__ Wm,
                           const float* __restrict__ bias,
                           float* __restrict__ query) {
  const int tile  = blockIdx.x;
  const int mBase = (tile & 1) * 16;     // 2 tiles in M (batch)
  const int nBase = (tile >> 1) * 16;    // 16 tiles in N (KS)
  const int lane  = threadIdx.x;         // wave32
  const int half  = (lane >> 4) & 1;
  const int l16   = lane & 15;

  const float* qrow = q  + (size_t)(mBase + l16) * QS_ + half * 2;
  const float* wrow = Wm + (size_t)(nBase + l16) * QS_ + half * 2;

  v8f acc = {};
  #pragma unroll 4
  for (int k0 = 0; k0 < QS_; k0 += 4) {
    v2f a = *(const v2f*)(qrow + k0);
    v2f b = *(const v2f*)(wrow + k0);
    // (neg_a, A, neg_b, B, c_mod, C, reuse_a, reuse_b)
    acc = __builtin_amdgcn_wmma_f32_16x16x4_f32(false, a, false, b,
                                                (short)0, acc, false, false);
  }

  const float bv = bias[nBase + l16];
  #pragma unroll
  for (int r = 0; r < 8; ++r) {
    const int m = mBase + r + half * 8;
    query[(size_t)m * KS_ + nBase + l16] = acc[r] + bv;
  }
}

// ---------------------------------------------------------------------------
// Kernel 2: attn scores k[b,n,:].query[b,:] then sparsemax over n (N=6).
// One wave32 per batch; butterfly reduction; lane 0 does the 6-elem sparsemax.
// ---------------------------------------------------------------------------
__global__ void attn_sparsemax(const float* __restrict__ kmat,
                               const float* __restrict__ query,
                               float* __restrict__ attn_ws,
                               float* __restrict__ out_attn) {
  const int b    = blockIdx.x;
  const int lane = threadIdx.x;   // 0..31

  float s[N_];
  #pragma unroll
  for (int n = 0; n < N_; ++n) s[n] = 0.0f;

  #pragma unroll
  for (int j = 0; j < KS_ / 32; ++j) {
    const int kk = lane + j * 32;
    const float qv = query[(size_t)b * KS_ + kk];
    #pragma unroll
    for (int n = 0; n < N_; ++n)
      s[n] += kmat[((size_t)b * N_ + n) * KS_ + kk] * qv;
  }

  #pragma unroll
  for (int n = 0; n < N_; ++n) {
    #pragma unroll
    for (int off = 16; off > 0; off >>= 1)
      s[n] += __shfl_xor(s[n], off, 32);
  }

  if (lane == 0) {
    float z[N_];
    #pragma unroll
    for (int n = 0; n < N_; ++n) z[n] = s[n];
    // sort descending (insertion sort, 6 elements)
    for (int i = 1; i < N_; ++i) {
      float key = z[i];
      int j = i - 1;
      while (j >= 0 && z[j] < key) { z[j + 1] = z[j]; --j; }
      z[j + 1] = key;
    }
    float cums[N_];
    float cum = 0.0f;
    #pragma unroll
    for (int r = 0; r < N_; ++r) { cum += z[r]; cums[r] = cum; }
    int ksup = 1;
    #pragma unroll
    for (int r = 1; r <= N_; ++r)
      if (1.0f + (float)r * z[r - 1] > cums[r - 1]) ksup = r;
    const float tau = (cums[ksup - 1] - 1.0f) / (float)ksup;
    #pragma unroll
    for (int n = 0; n < N_; ++n) {
      float p = s[n] - tau;
      p = p > 0.0f ? p : 0.0f;
      attn_ws[b * N_ + n]  = p;   // for kernel 3
      out_attn[b * N_ + n] = p;   // second output: attn^T [B,1,N] flat
    }
  }
}

// ---------------------------------------------------------------------------
// Kernel 3: out[b,i] = sum_n attn[b,n] * v[b,n,i]  (i over CHW) — the
// bandwidth kernel: ~402 MB read + 67 MB write, all b128 coalesced.
// ---------------------------------------------------------------------------
__global__ void weighted_sum(const float* __restrict__ v,
                             const float* __restrict__ attn,
                             float* __restrict__ out) {
  const int b   = blockIdx.x / BLOCKS_PER_BATCH;
  const int blk = blockIdx.x % BLOCKS_PER_BATCH;

  __shared__ float a_s[N_];
  if (threadIdx.x < N_) a_s[threadIdx.x] = attn[b * N_ + threadIdx.x];
  __syncthreads();

  const float4* vb = (const float4*)v + (size_t)b * N_ * F4_PER_BATCH;
  float4*       ob = (float4*)out     + (size_t)b * F4_PER_BATCH;

  const int base = blk * 512 + threadIdx.x;
  #pragma unroll
  for (int it = 0; it < 2; ++it) {
    const int i = base + it * 256;
    float4 acc = make_float4(0.0f, 0.0f, 0.0f, 0.0f);
    #pragma unroll
    for (int n = 0; n < N_; ++n) {
      const float4 x = vb[(size_t)n * F4_PER_BATCH + i];
      const float  a = a_s[n];
      acc.x += a * x.x; acc.y += a * x.y;
      acc.z += a * x.z; acc.w += a * x.w;
    }
    ob[i] = acc;
  }
}

extern "C" void kernel_launch(void* const* d_in, const int* in_sizes, int n_in,
                              void* d_out, int out_size, void* d_ws, size_t ws_size,
                              hipStream_t stream) {
  const float* q    = (const float*)d_in[0];  // [32,1,256]
  const float* kmat = (const float*)d_in[1];  // [32,6,256]
  const float* v    = (const float*)d_in[2];  // [32,6,512,32,32]
  const float* Wm   = (const float*)d_in[3];  // [256,256]
  const float* bias = (const float*)d_in[4];  // [256]

  float* out      = (float*)d_out;                          // [32,512,32,32]
  float* out_attn = out + (size_t)B_ * C_ * H_ * W_;        // [32,1,6]

  float* query_ws = (float*)d_ws;            // 32*256 floats
  float* attn_ws  = query_ws + B_ * KS_;     // 32*6 floats

  qproj_wmma<<<32, 32, 0, stream>>>(q, Wm, bias, query_ws);
  attn_sparsemax<<<B_, 32, 0, stream>>>(kmat, query_ws, attn_ws, out_attn);
  weighted_sum<<<B_ * BLOCKS_PER_BATCH, 256, 0, stream>>>(v, attn_ws, out);
}